// TopologicalQualiaLoss_8358006358178
// MI455X (gfx1250) — compile-verified
//
#include <hip/hip_runtime.h>

typedef __attribute__((ext_vector_type(2))) float v2f;
typedef __attribute__((ext_vector_type(8))) float v8f;

#define T 8192
#define D 256
#define KNN 5
#define NCHUNK 8
#define CHUNK 1024
#define TILES_PER_CHUNK (CHUNK / 16)

// Branch-free sorted insertion: b[] ascending, keep 5 smallest.
__device__ __forceinline__ void ins5(float v, float b[KNN]) {
  float x = v;
  #pragma unroll
  for (int i = 0; i < KNN; ++i) {
    float t = fminf(b[i], x);
    x = fmaxf(b[i], x);
    b[i] = t;
  }
}

// Async-copy one 16x256 f32 panel (16 KB) global -> LDS using
// GLOBAL_LOAD_ASYNC_TO_LDS_B128 (ASYNCcnt path). 32 ops per wave.
// ldsBase: LDS byte offset of panel start; gsrc: uniform global base.
__device__ __forceinline__ void async_panel(unsigned ldsBase,
                                            const float* gsrc, int lane) {
  unsigned l = ldsBase + (unsigned)lane * 16u;
  unsigned off = (unsigned)lane * 16u;
  #pragma unroll
  for (int i = 0; i < 32; ++i) {
    asm volatile("global_load_async_to_lds_b128 %0, %1, %2"
                 :: "v"(l), "v"(off), "s"(gsrc)
                 : "memory");
    l += 512u;
    off += 512u;
  }
}

__device__ __forceinline__ void wait_async_le32() {
  asm volatile("s_wait_asynccnt 0x20" ::: "memory");
}
__device__ __forceinline__ void wait_async_0() {
  asm volatile("s_wait_asynccnt 0x0" ::: "memory");
}

// ---------------- Kernel 0: row squared norms ----------------
__global__ void __launch_bounds__(256) k_sqnorm(const float* __restrict__ S,
                                                float* __restrict__ sq) {
  int row = blockIdx.x * blockDim.x + threadIdx.x;
  const float4* p = (const float4*)(S + (size_t)row * D);
  float s = 0.f;
  #pragma unroll 8
  for (int i = 0; i < D / 4; ++i) {
    float4 v = p[i];
    s += v.x * v.x + v.y * v.y + v.z * v.z + v.w * v.w;
  }
  sq[row] = s;
}

// ---------------- Kernel 1: fused WMMA Gram-tile + top-5 ----------------
// grid = (T/16, NCHUNK), block = 32 (one wave32).
// Each wave: rows [16*bx, 16*bx+16) x cols [1024*by, 1024*by+1024).
// B panels are double-buffered via async global->LDS copies.
__global__ void __launch_bounds__(32)
k_top5_partial(const float* __restrict__ S, const float* __restrict__ sq,
               float* __restrict__ part) {
  __shared__ __align__(16) float Asub[16 * D];        // 16 KB
  __shared__ __align__(16) float Bsub[2][16 * D];     // 32 KB
  __shared__ float Merge[16][16 * KNN];               // 5 KB

  const int lane = threadIdx.x;
  const int g = lane >> 4;        // lane group: rows 8g..8g+7 in C tile
  const int lm = lane & 15;       // column within tile; row M for A/B frags
  const int rowBase = blockIdx.x * 16;
  const int colBase = blockIdx.y * CHUNK;

  const unsigned ldsA  = (unsigned)(size_t)&Asub[0];
  const unsigned ldsB0 = (unsigned)(size_t)&Bsub[0][0];
  const unsigned ldsB1 = (unsigned)(size_t)&Bsub[1][0];

  // Kick off async copies: A panel, then B panel 0 (in-order completion).
  async_panel(ldsA, S + (size_t)rowBase * D, lane);
  async_panel(ldsB0, S + (size_t)colBase * D, lane);

  // Row squared norms for the 8 rows this lane owns in the C tile
  // (separate LOADcnt path, overlaps with the async copies).
  float rowSq[8];
  #pragma unroll
  for (int r = 0; r < 8; ++r) rowSq[r] = sq[rowBase + 8 * g + r];

  float best[8][KNN];
  #pragma unroll
  for (int r = 0; r < 8; ++r)
    #pragma unroll
    for (int q = 0; q < KNN; ++q) best[r][q] = 3.0e38f;

  for (int jt = 0; jt < TILES_PER_CHUNK; ++jt) {
    const int col0 = colBase + jt * 16;

    if (jt + 1 < TILES_PER_CHUNK) {
      // Prefetch next B panel into the other buffer, then wait until only
      // those 32 ops remain in flight => A + current B have landed.
      async_panel(((jt + 1) & 1) ? ldsB1 : ldsB0,
                  S + (size_t)(col0 + 16) * D, lane);
      wait_async_le32();
    } else {
      wait_async_0();
    }

    // A frag (16x4 f32): lane m=lm, VGPR0/1 = K = 4k + 2g + {0,1}.
    // B frag (4x16 f32) = S^T tile: identical per-lane addressing.
    v8f acc = {};
    const float* ap = Asub + lm * D + 2 * g;
    const float* bp = Bsub[jt & 1] + lm * D + 2 * g;
    #pragma unroll 8
    for (int k = 0; k < D / 4; ++k) {
      v2f a = *(const v2f*)(ap + 4 * k);
      v2f b = *(const v2f*)(bp + 4 * k);
      acc = __builtin_amdgcn_wmma_f32_16x16x4_f32(
          false, a, false, b, (short)0, acc, false, false);
    }

    // d2 = sq_i + sq_j - 2*dot, clamp >= 0, insert into per-row top-5.
    const float csq = sq[col0 + lm];
    #pragma unroll
    for (int r = 0; r < 8; ++r) {
      float v = fmaf(-2.0f, acc[r], rowSq[r] + csq);
      v = fmaxf(v, 0.0f);
      ins5(v, best[r]);
    }
  }

  // Merge the 16 per-lane lists for each row (each row's columns are
  // striped across the 16 lanes of one lane-group).
  #pragma unroll
  for (int r = 0; r < 8; ++r)
    #pragma unroll
    for (int q = 0; q < KNN; ++q)
      Merge[8 * g + r][lm * KNN + q] = best[r][q];
  __syncthreads();

  if (lane < 16) {
    float out5[KNN];
    #pragma unroll
    for (int q = 0; q < KNN; ++q) out5[q] = 3.0e38f;
    for (int i = 0; i < 16 * KNN; ++i) ins5(Merge[lane][i], out5);
    float* dst = part + (size_t)(rowBase + lane) * (NCHUNK * KNN)
                      + (size_t)blockIdx.y * KNN;
    #pragma unroll
    for (int q = 0; q < KNN; ++q) dst[q] = out5[q];
  }
}

// ---------------- Kernel 2: merge chunks, sqrt, accumulate stats ----------------
__global__ void __launch_bounds__(256)
k_merge_stats(const float* __restrict__ part, float* __restrict__ accum) {
  __shared__ float rs[256];
  __shared__ float rq[256];
  const int tid = threadIdx.x;
  const int row = blockIdx.x * 256 + tid;

  const float* p = part + (size_t)row * (NCHUNK * KNN);
  float out5[KNN];
  #pragma unroll
  for (int q = 0; q < KNN; ++q) out5[q] = 3.0e38f;
  #pragma unroll
  for (int i = 0; i < NCHUNK * KNN; ++i) ins5(p[i], out5);

  float s1 = 0.f, s2 = 0.f;
  #pragma unroll
  for (int q = 0; q < KNN; ++q) {
    float d2 = out5[q];
    float dist = (d2 > 0.f) ? sqrtf(d2) : 0.f;  // safe sqrt, matches reference
    s1 += dist;
    s2 += dist * dist;
  }

  rs[tid] = s1; rq[tid] = s2;
  __syncthreads();
  for (int off = 128; off > 0; off >>= 1) {
    if (tid < off) { rs[tid] += rs[tid + off]; rq[tid] += rq[tid + off]; }
    __syncthreads();
  }
  if (tid == 0) {
    atomicAdd(&accum[0], rs[0]);
    atomicAdd(&accum[1], rq[0]);
  }
}

// ---------------- Kernel 3: finalize -std ----------------
__global__ void k_finalize(const float* __restrict__ accum,
                           float* __restrict__ out) {
  float s1 = accum[0], s2 = accum[1];
  const float n = (float)(T * KNN);
  float ssd = s2 - s1 * s1 / n;     // sum((x-mean)^2)
  ssd = fmaxf(ssd, 0.f);
  out[0] = -sqrtf(ssd / (n - 1.0f));
}

extern "C" void kernel_launch(void* const* d_in, const int* in_sizes, int n_in,
                              void* d_out, int out_size, void* d_ws, size_t ws_size,
                              hipStream_t stream) {
  (void)in_sizes; (void)n_in; (void)out_size; (void)ws_size;
  const float* latent = (const float*)d_in[0];
  const float* S = latent;                       // sample = latent[0]
  char* ws = (char*)d_ws;
  float* accum = (float*)ws;                                     // 2 f32 (64 B slot)
  float* sq    = (float*)(ws + 64);                              // T f32
  float* part  = (float*)(ws + 64 + (size_t)T * sizeof(float));  // T*NCHUNK*KNN f32
  float* out = (float*)d_out;

  hipMemsetAsync(accum, 0, 64, stream);
  k_sqnorm<<<T / 256, 256, 0, stream>>>(S, sq);
  dim3 grid1(T / 16, NCHUNK);
  k_top5_partial<<<grid1, 32, 0, stream>>>(S, sq, part);
  k_merge_stats<<<T / 256, 256, 0, stream>>>(part, accum);
  k_finalize<<<1, 1, 0, stream>>>(accum, out);
}